// ChannelAssembly_6270652252219
// MI455X (gfx1250) — compile-verified
//
#include <hip/hip_runtime.h>
#include <hip/hip_bf16.h>

typedef __attribute__((ext_vector_type(2))) float v2f;
typedef __attribute__((ext_vector_type(8))) float v8f;

#define C_CH    192     // channels
#define T_MERGE 96      // C - original_channels
#define BATCH   32
#define HW      3136    // 56*56
#define HW4     784     // HW / 4 (float4)
#define NT      12      // 192/16 tiles per matrix dim
#define NTILES  144     // NT*NT
#define KSLICES 16      // K split: each slice = 2 batches = 6272 K values
#define BPS     2       // batches per K-slice

// ---------------------------------------------------------------------------
// Kernel 1: partial Gram tiles via V_WMMA_F32_16X16X4_F32.
// One wave (32 threads) per (tile, kslice). A and B fragments of the Gram
// share the identical lane layout: lane L holds row base + (L%16), K-offset
// 2*(L/16), two consecutive f32 per VGPR pair -> 8-byte global_load_b64.
// ---------------------------------------------------------------------------
__global__ __launch_bounds__(32)
void gram_wmma_kernel(const float* __restrict__ x, float* __restrict__ Gpart) {
    const int tile = blockIdx.x;            // 0..143
    const int ti   = tile / NT;
    const int tj   = tile % NT;
    const int ks   = blockIdx.y;            // 0..15
    const int lane = threadIdx.x;           // 0..31
    const int mrow = lane & 15;
    const int koff = (lane >> 4) << 1;      // 0 or 2
    const int row_a = ti * 16 + mrow;
    const int row_b = tj * 16 + mrow;

    v8f acc = {};
    for (int bb = ks * BPS; bb < ks * BPS + BPS; ++bb) {
        const float* pa = x + ((size_t)bb * C_CH + row_a) * HW + koff;
        const float* pb = x + ((size_t)bb * C_CH + row_b) * HW + koff;
        for (int r = 0; r < HW; r += 16) {
#pragma unroll
            for (int u = 0; u < 4; ++u) {
                v2f av = *(const v2f*)(pa + r + 4 * u);
                v2f bv = *(const v2f*)(pb + r + 4 * u);
                // D = A x B + C  (16x16x4, f32 in, f32 acc)
                acc = __builtin_amdgcn_wmma_f32_16x16x4_f32(
                    /*neg_a=*/false, av, /*neg_b=*/false, bv,
                    /*c_mod=*/(short)0, acc, /*reuse_a=*/false, /*reuse_b=*/false);
            }
        }
    }
    // lane-major store: element (m,n) of tile lives at lane*8 + vgpr,
    // with lane = n + 16*(m/8), vgpr = m%8  (per CDNA5 16x16 f32 C/D layout).
    float* outp = Gpart + ((size_t)tile * KSLICES + ks) * 256 + lane * 8;
#pragma unroll
    for (int r = 0; r < 8; ++r) outp[r] = acc[r];
}

// ---------------------------------------------------------------------------
// Kernel 2: reduce K-slices + form D2[i,j] = G[i,i] + G[j,j] - 2*G[i,j].
// Fixed summation order -> bit-deterministic across replays.
// ---------------------------------------------------------------------------
__device__ __forceinline__ float tile_sum(const float* __restrict__ Gpart,
                                          int i, int j) {
    const int tile = (i >> 4) * NT + (j >> 4);
    const int m = i & 15, n = j & 15;
    const int e = (n + 16 * (m >> 3)) * 8 + (m & 7);
    const float* p = Gpart + (size_t)tile * KSLICES * 256 + e;
    float s = 0.0f;
#pragma unroll
    for (int sl = 0; sl < KSLICES; ++sl) s += p[sl * 256];
    return s;
}

__global__ void reduce_d2_kernel(const float* __restrict__ Gpart,
                                 float* __restrict__ D2) {
    const int id = blockIdx.x * blockDim.x + threadIdx.x;
    if (id >= C_CH * C_CH) return;
    const int i = id / C_CH, j = id % C_CH;
    const float gij = tile_sum(Gpart, i, j);
    const float gii = tile_sum(Gpart, i, i);
    const float gjj = tile_sum(Gpart, j, j);
    D2[id] = gii + gjj - 2.0f * gij;
}

// ---------------------------------------------------------------------------
// Kernel 3: greedy merge loop (single workgroup). Key = (f32 bits << 32) | l
// with l = i*192 + j: unsigned min == first-occurrence row-major argmin
// (distances are >= 0 so f32 bit patterns are monotone as u32).
// Replicates the reference's "bug": D2 is never updated; appended index is t.
// ---------------------------------------------------------------------------
__global__ __launch_bounds__(256)
void merge_kernel(const float* __restrict__ D2, int* __restrict__ pairs) {
    __shared__ int idx[C_CH];
    __shared__ unsigned long long best;
    const int tid = threadIdx.x;
    for (int c = tid; c < C_CH; c += 256) idx[c] = c;
    __syncthreads();

    for (int t = 0; t < T_MERGE; ++t) {
        const int n = C_CH - t;
        if (tid == 0) best = ~0ull;
        __syncthreads();

        unsigned long long loc = ~0ull;
        for (int l = tid; l < C_CH * C_CH; l += 256) {
            const int i = l / C_CH, j = l % C_CH;
            if (i < j && j < n) {
                const float d = D2[idx[i] * C_CH + idx[j]];
                const unsigned long long key =
                    ((unsigned long long)__float_as_uint(d) << 32) |
                    (unsigned int)l;
                loc = (key < loc) ? key : loc;
            }
        }
        atomicMin(&best, loc);   // ds_min_u64 on LDS
        __syncthreads();

        if (tid == 0) {
            const unsigned int l = (unsigned int)(best & 0xFFFFFFFFu);
            const int i = (int)(l / C_CH), j = (int)(l % C_CH);
            pairs[2 * t]     = idx[i];
            pairs[2 * t + 1] = idx[j];
            // pop position j, then position i (i < j), append t
            for (int p = j; p < n - 1; ++p) idx[p] = idx[p + 1];
            for (int p = i; p < n - 2; ++p) idx[p] = idx[p + 1];
            idx[n - 2] = t;
        }
        __syncthreads();
    }
}

// ---------------------------------------------------------------------------
// Kernel 4: out[b,t,:,:] = 0.5*(x[b,v1,:,:] + x[b,v2,:,:]), float4 vectorized.
// ---------------------------------------------------------------------------
__global__ __launch_bounds__(256)
void assemble_kernel(const float* __restrict__ x,
                     const int* __restrict__ pairs,
                     float* __restrict__ out) {
    const int id = blockIdx.x * blockDim.x + threadIdx.x;
    if (id >= BATCH * T_MERGE * HW4) return;
    const int hw = id % HW4;
    const int t  = (id / HW4) % T_MERGE;
    const int b  = id / (HW4 * T_MERGE);
    const int v1 = pairs[2 * t];
    const int v2 = pairs[2 * t + 1];
    const float4* xv = (const float4*)x;
    const float4 a = xv[((size_t)b * C_CH + v1) * HW4 + hw];
    const float4 c = xv[((size_t)b * C_CH + v2) * HW4 + hw];
    float4 r;
    r.x = 0.5f * (a.x + c.x);
    r.y = 0.5f * (a.y + c.y);
    r.z = 0.5f * (a.z + c.z);
    r.w = 0.5f * (a.w + c.w);
    ((float4*)out)[id] = r;
}

// ---------------------------------------------------------------------------
extern "C" void kernel_launch(void* const* d_in, const int* in_sizes, int n_in,
                              void* d_out, int out_size, void* d_ws, size_t ws_size,
                              hipStream_t stream) {
    const float* x = (const float*)d_in[0];
    float* out = (float*)d_out;

    // workspace layout (floats): Gpart[144*16*256] | D2[192*192] | pairs[192 ints]
    float* Gpart = (float*)d_ws;
    float* D2    = Gpart + (size_t)NTILES * KSLICES * 256;
    int*   pairs = (int*)(D2 + C_CH * C_CH);

    gram_wmma_kernel<<<dim3(NTILES, KSLICES), 32, 0, stream>>>(x, Gpart);

    const int nd2 = C_CH * C_CH;
    reduce_d2_kernel<<<(nd2 + 255) / 256, 256, 0, stream>>>(Gpart, D2);

    merge_kernel<<<1, 256, 0, stream>>>(D2, pairs);

    const int nout4 = BATCH * T_MERGE * HW4;
    assemble_kernel<<<(nout4 + 255) / 256, 256, 0, stream>>>(x, pairs, out);
}